// MacLaurinRegression_74835510165593
// MI455X (gfx1250) — compile-verified
//
#include <hip/hip_runtime.h>

typedef __attribute__((ext_vector_type(2))) float v2f;
typedef __attribute__((ext_vector_type(4))) float v4f;
typedef __attribute__((ext_vector_type(8))) float v8f;

#define BATCH   65536
#define FEAT    8
#define T_REAL  495
#define T_PAD   496          // 31 tiles of 16 columns
#define N_TILES 31
#define BLOCK   256          // 8 waves (wave32)
#define WPB     8

__global__ __launch_bounds__(BLOCK) void maclaurin_wmma_f32(
    const float* __restrict__ x,      // [BATCH, 8]
    const float* __restrict__ terms,  // [495, 8]
    const float* __restrict__ coefs,  // [495]
    const float* __restrict__ facs,   // [495]
    float* __restrict__ out)          // [BATCH]
{
    __shared__ float sTerms[T_PAD * FEAT]; // exponent table, padded row 495 = 0
    __shared__ float sW[T_PAD];            // facs*coefs, padded col 495 = 0

    const int tid = threadIdx.x;
    for (int i = tid; i < T_PAD * FEAT; i += BLOCK)
        sTerms[i] = (i < T_REAL * FEAT) ? terms[i] : 0.0f;
    for (int i = tid; i < T_PAD; i += BLOCK)
        sW[i] = (i < T_REAL) ? facs[i] * coefs[i] : 0.0f;
    __syncthreads();

    const int lane   = tid & 31;
    const int laneLo = lane & 15;     // M (for A/C) or N (for B) index
    const int hi     = lane >> 4;     // selects K-pair / M-half per ISA layout
    const int waveId = blockIdx.x * WPB + (tid >> 5);
    const int rowBase = waveId * 16;  // 16 batch rows per wave

    // ---- A matrix: log2(x) for 16 rows, K=8 split into two 16x4 operands ----
    // VGPR0 holds K=2*hi, VGPR1 holds K=2*hi+1 (ISA 32-bit A layout).
    const float* xr = x + (rowBase + laneLo) * FEAT;
    v2f x0 = *(const v2f*)(xr + 2 * hi);
    v2f x1 = *(const v2f*)(xr + 2 * hi + 4);
    v2f a_lo, a_hi;
    a_lo.x = __builtin_amdgcn_logf(x0.x);   // v_log_f32 (log2)
    a_lo.y = __builtin_amdgcn_logf(x0.y);
    a_hi.x = __builtin_amdgcn_logf(x1.x);
    a_hi.y = __builtin_amdgcn_logf(x1.y);

    float acc[8];
#pragma unroll
    for (int r = 0; r < 8; ++r) acc[r] = 0.0f;

    for (int n = 0; n < N_TILES; ++n) {
        const int t = n * 16 + laneLo;            // global column (term index)
        const float* tr = &sTerms[t * FEAT];
        v2f b_lo = *(const v2f*)(tr + 2 * hi);    // B[k=2*hi..+1][N=laneLo]
        v2f b_hi = *(const v2f*)(tr + 2 * hi + 4);
        const float wv = sW[t];

        // D = A*B + C, chained over K=8, full f32 precision
        v8f c = {};
        c = __builtin_amdgcn_wmma_f32_16x16x4_f32(false, a_lo, false, b_lo,
                                                  (short)0, c, false, false);
        c = __builtin_amdgcn_wmma_f32_16x16x4_f32(false, a_hi, false, b_hi,
                                                  (short)0, c, false, false);

        // mono = 2^(E . log2 x) = prod x^e ; weighted accumulate per row
#pragma unroll
        for (int r = 0; r < 8; ++r)
            acc[r] = fmaf(wv, __builtin_amdgcn_exp2f(c[r]), acc[r]);
    }

    // ---- reduce over the 16 columns held across lanes of each half-wave ----
#pragma unroll
    for (int r = 0; r < 8; ++r) {
        acc[r] += __shfl_xor(acc[r], 1, 32);
        acc[r] += __shfl_xor(acc[r], 2, 32);
        acc[r] += __shfl_xor(acc[r], 4, 32);
        acc[r] += __shfl_xor(acc[r], 8, 32);
    }

    // lane 0 holds rows M=0..7, lane 16 holds rows M=8..15 (C layout)
    if (laneLo == 0) {
        float* o = out + rowBase + hi * 8;
        v4f o0 = {acc[0], acc[1], acc[2], acc[3]};
        v4f o1 = {acc[4], acc[5], acc[6], acc[7]};
        *(v4f*)(o)     = o0;   // two coalesced b128 stores
        *(v4f*)(o + 4) = o1;
    }
}

extern "C" void kernel_launch(void* const* d_in, const int* in_sizes, int n_in,
                              void* d_out, int out_size, void* d_ws, size_t ws_size,
                              hipStream_t stream) {
    (void)in_sizes; (void)n_in; (void)d_ws; (void)ws_size; (void)out_size;
    const float* x     = (const float*)d_in[0];
    const float* terms = (const float*)d_in[1];
    const float* coefs = (const float*)d_in[2];
    const float* facs  = (const float*)d_in[3];
    float* out = (float*)d_out;

    const int rowsPerBlock = 16 * WPB;              // 128 rows per block
    const int grid = BATCH / rowsPerBlock;          // 512 blocks
    maclaurin_wmma_f32<<<grid, BLOCK, 0, stream>>>(x, terms, coefs, facs, out);
}